// FoE_11519102288693
// MI455X (gfx1250) — compile-verified
//
#include <hip/hip_runtime.h>
#include <math.h>

typedef __attribute__((ext_vector_type(2))) float v2f;
typedef __attribute__((ext_vector_type(8))) float v8f;

#define HH 256
#define WW 256
#define NB 4
#define NF 16
#define NC 20
#define NPIX (HH * WW)
#define TROWS 14   // 8 output rows + 2*3 halo
#define TCOLS 32   // 16 output cols + halo, padded to 32 for shift/and indexing
#define KPAD 56    // 7 rows x 2 quads x 4 = 56 (taps reordered into row-quads)

// ---------------- per-image mean / clamped inv-std ----------------
__global__ void foe_stats_kernel(const float* __restrict__ src, float* __restrict__ stats) {
    __shared__ float s_sum[256];
    __shared__ float s_sq[256];
    const int b = blockIdx.x;
    const float* p = src + b * NPIX;
    float sum = 0.f, sq = 0.f;
    for (int i = threadIdx.x; i < NPIX; i += 256) {
        float v = p[i];
        sum += v;
        sq += v * v;
    }
    s_sum[threadIdx.x] = sum;
    s_sq[threadIdx.x] = sq;
    __syncthreads();
    for (int off = 128; off > 0; off >>= 1) {
        if (threadIdx.x < off) {
            s_sum[threadIdx.x] += s_sum[threadIdx.x + off];
            s_sq[threadIdx.x]  += s_sq[threadIdx.x + off];
        }
        __syncthreads();
    }
    if (threadIdx.x == 0) {
        float mean = s_sum[0] * (1.0f / (float)NPIX);
        float var  = fmaxf(s_sq[0] * (1.0f / (float)NPIX) - mean * mean, 0.f);
        float sd   = sqrtf(var);
        float den  = fmaxf(sd, 1.0f / 256.0f);   // 1/sqrt(65536)
        stats[b * 2 + 0] = mean;
        stats[b * 2 + 1] = 1.0f / den;
    }
}

// ------- fused conv (WMMA, filters on M / pixels on N) + sigmoid + diff-op -> g -------
// K-quad s covers tile positions (s/2)*TCOLS + (s&1)*4 + {0,1,2,3}; weights are
// permuted to match at fill time, with zeros at the dj==7 pad positions.
__global__ void __launch_bounds__(256)
foe_conv_g_kernel(const float* __restrict__ src, const float* __restrict__ Wexp,
                  const float* __restrict__ Wf, const float* __restrict__ bg,
                  const float* __restrict__ stats, float* __restrict__ gout) {
    __shared__ float s_tile[TROWS * TCOLS];   // raw pixels, OOB -> mean (z=0)
    __shared__ float s_w[KPAD * NF];          // quad-ordered taps x filters, zero padded
    __shared__ float s_f[NF * NC];            // W_f rows
    __shared__ float s_pf[NF * (NC + 1)];     // per-filter prefix sums of W_f
    __shared__ float s_sw[NF];                // per-filter sum of conv taps
    __shared__ float s_bg2sum;                // sum_m b_g[m]^2

    // block -> (b, row-block of 8, col-block of 16)
    const int bid = blockIdx.x;
    const int cb  = bid & 15;            // WW/16 = 16
    const int rb  = (bid >> 4) & 31;     // HH/8  = 32
    const int b   = bid >> 9;
    const int i0  = rb * 8;
    const int j0  = cb * 16;

    const float mean   = stats[b * 2 + 0];
    const float invstd = stats[b * 2 + 1];
    const float* img   = src + b * NPIX;

    const int tid = threadIdx.x;
    // weights, permuted into row-quads: k -> (di = k/8, dj = ((k/4)&1)*4 + (k&3))
    for (int idx = tid; idx < KPAD * NF; idx += 256) {
        int k = idx >> 4, n = idx & 15;
        int di = k >> 3, dj = ((k >> 2) & 1) * 4 + (k & 3);
        s_w[idx] = (dj < 7) ? Wexp[(di * 7 + dj) * NF + n] : 0.f;
    }
    for (int idx = tid; idx < NF * NC; idx += 256)
        s_f[idx] = Wf[idx];
    if (tid < NF) {
        float a = 0.f;
        for (int c = 0; c < NC; ++c) { s_pf[tid * (NC + 1) + c] = a; a += Wf[tid * NC + c]; }
        s_pf[tid * (NC + 1) + NC] = a;
        float sw = 0.f;
        for (int t = 0; t < 49; ++t) sw += Wexp[t * NF + tid];
        s_sw[tid] = sw;
    }
    if (tid == 16) {
        float s = 0.f;
        for (int n = 0; n < NF; ++n) { float v = bg[n]; s += v * v; }
        s_bg2sum = s;
    }
    for (int idx = tid; idx < TROWS * TCOLS; idx += 256) {
        int r = idx >> 5, c = idx & 31;
        int gr = i0 - 3 + r, gc = j0 - 3 + c;
        float v = mean;                                  // z = 0 after standardization
        if (gr >= 0 && gr < HH && gc >= 0 && gc < WW) v = img[gr * WW + gc];
        s_tile[idx] = v;
    }
    __syncthreads();

    const int  wave = tid >> 5;        // 0..7 -> output row i0+wave
    const int  lid  = tid & 31;
    const int  ln   = lid & 15;        // A: filter index / B & D: pixel index
    const bool hi   = (lid >= 16);     // lanes 16-31 hold K-quad elements 2,3
    const int  half = hi ? 1 : 0;

    // half-dependent parts folded into the bases once; loop offsets are immediates
    const float* wb = s_w + (hi ? 2 * NF : 0) + ln;
    const float* tb = s_tile + wave * TCOLS + ln + (hi ? 2 : 0);

    v8f acc = {0.f, 0.f, 0.f, 0.f, 0.f, 0.f, 0.f, 0.f};
#pragma unroll
    for (int s = 0; s < 14; ++s) {
        const int to = (s >> 1) * TCOLS + (s & 1) * 4;   // compile-time quad origin
        v2f a, bf;
        a.x  = wb[s * 4 * NF];           // A = weights (filters on M)
        a.y  = wb[s * 4 * NF + NF];
        bf.x = tb[to];                   // B = pixels, consecutive -> immediate offsets
        bf.y = tb[to + 1];
        acc = __builtin_amdgcn_wmma_f32_16x16x4_f32(
            false, a, false, bf, (short)0, acc, false, false);
    }

    // D: register r, this lane -> (filter m = r + 8*half, pixel n = ln)
    float swv[8];
#pragma unroll
    for (int r = 0; r < 8; ++r) swv[r] = s_sw[half * 8 + r];   // consecutive -> wide DS loads
    const float* pfb = s_pf + half * 8 * (NC + 1);
    const float* fb  = s_f  + half * 8 * NC;

    float fsum = 0.f;
#pragma unroll
    for (int r = 0; r < 8; ++r) {
        float y = (acc[r] - mean * swv[r]) * invstd;   // conv linearity correction
        float e = 1.0f / (1.0f + expf(-y));            // sigmoid, e in (0,1]
        int cbin = (int)ceilf(e * 20.0f) - 1;          // bin with e in (c/20,(c+1)/20]
        cbin = cbin < 0 ? 0 : (cbin > 19 ? 19 : cbin);
        float mod = fmodf(e, 0.05f);
        float bias = pfb[r * (NC + 1) + cbin];         // sum of full bins below
        float fsel = fb[r * NC + cbin];                // f[m][bin]
        fsum += bias * 0.05f + fsel * mod;             // h = f/C
    }
    // add the other half's 8 filters, then the bg^2 block constant
    fsum += __shfl_xor(fsum, 16, 32);
    fsum += s_bg2sum;
    float g = expf(fsum);

    const int i = i0 + wave;
    if (lid < 16) gout[(b * HH + i) * WW + j0 + ln] = g;
}

// ---------------- divergence update: out += gamma*(dS/dy + dE/dx) ----------------
__global__ void foe_update_kernel(const float* __restrict__ src, const float* __restrict__ g,
                                  float* __restrict__ dst) {
    int idx = blockIdx.x * 256 + threadIdx.x;
    if (idx >= NB * NPIX) return;
    int j = idx & (WW - 1);
    int t = idx >> 8;
    int i = t & (HH - 1);
    int b = t >> 8;
    const float* im = src + b * NPIX;
    const float* gb = g + b * NPIX;
    float x = im[i * WW + j];
    float Sij = (i < HH - 1) ? gb[i * WW + j] * (im[(i + 1) * WW + j] - x) : 0.f;
    float Sup = (i > 0)      ? gb[(i - 1) * WW + j] * (x - im[(i - 1) * WW + j]) : 0.f;
    float Eij = (j < WW - 1) ? gb[i * WW + j] * (im[i * WW + j + 1] - x) : 0.f;
    float Elf = (j > 0)      ? gb[i * WW + j - 1] * (x - im[i * WW + j - 1]) : 0.f;
    dst[idx] = x + 0.1f * ((Sij - Sup) + (Eij - Elf));
}

extern "C" void kernel_launch(void* const* d_in, const int* in_sizes, int n_in,
                              void* d_out, int out_size, void* d_ws, size_t ws_size,
                              hipStream_t stream) {
    (void)in_sizes; (void)n_in; (void)out_size; (void)ws_size;
    const float* x    = (const float*)d_in[0];
    const float* Wexp = (const float*)d_in[1];
    const float* Wf   = (const float*)d_in[2];
    const float* bg   = (const float*)d_in[3];
    float* out   = (float*)d_out;
    float* buf0  = (float*)d_ws;
    float* buf1  = buf0 + NB * NPIX;
    float* gbuf  = buf1 + NB * NPIX;
    float* stats = gbuf + NB * NPIX;

    const float* cur = x;
    for (int it = 0; it < 10; ++it) {
        float* dst = (it == 9) ? out : ((it & 1) ? buf1 : buf0);
        foe_stats_kernel<<<NB, 256, 0, stream>>>(cur, stats);
        foe_conv_g_kernel<<<NB * (HH / 8) * (WW / 16), 256, 0, stream>>>(
            cur, Wexp, Wf, bg, stats, gbuf);
        foe_update_kernel<<<(NB * NPIX) / 256, 256, 0, stream>>>(cur, gbuf, dst);
        cur = dst;
    }
}